// OutputLayer_31791347925878
// MI455X (gfx1250) — compile-verified
//
#include <hip/hip_runtime.h>
#include <hip/hip_bf16.h>

typedef __attribute__((ext_vector_type(2))) float v2f;
typedef __attribute__((ext_vector_type(8))) float v8f;

#define N_NODES 20000
#define F_DIM   128
#define R_DIM   16
#define N_LAYER 3
#define O_DIM   12

// ---------------------------------------------------------------------------
// Kernel 0: zero the node accumulator in workspace
// ---------------------------------------------------------------------------
__global__ void zero_acc_kernel(float* __restrict__ p, long n) {
    long i = (long)blockIdx.x * blockDim.x + threadIdx.x;
    if (i < n) p[i] = 0.0f;
}

// ---------------------------------------------------------------------------
// Kernel 1: e_ji = rbf @ W_rbf^T via V_WMMA_F32_16X16X4_F32,
//           gate by m_ji, scatter-add to nodes with global_atomic_add_f32.
// One wave per 16-edge tile; 8 N-tiles of 16 features; K=16 as 4 wmma steps.
// m_ji / rbf are streamed non-temporally (read-once, 370 MB) so the
// L2-resident 10 MB node accumulator and MLP weights stay cached.
// The 8 m_ji gate loads per feature tile are issued as one batch (clause)
// before the multiply+atomic burst, so their latency overlaps.
// ---------------------------------------------------------------------------
__global__ __launch_bounds__(256) void edge_gemm_scatter_kernel(
    const float* __restrict__ m_ji,     // [E,128]
    const float* __restrict__ rbf,      // [E,16]
    const int*   __restrict__ eidx,     // [2,E] (row 1 = receiver)
    const float* __restrict__ W_rbf,    // [128,16]
    float*       __restrict__ acc,      // [N_NODES,128]
    int E)
{
    const int lane = threadIdx.x & 31;
    const int wave = threadIdx.x >> 5;
    const int l16  = lane & 15;
    const int half = lane >> 4;

    const long tile = (long)blockIdx.x * 8 + wave;  // 16-edge tile id
    const long e0   = tile * 16;
    if (e0 + 16 > E) return;

    const int* idx1 = eidx + E;  // receiving-atom row

    // A fragments: A[M=edge, K=rbf].  Lane layout (16x4 f32 A):
    //   lanes 0-15: row=l16, v0=K0, v1=K1 ; lanes 16-31: row=l16, v0=K2, v1=K3
    v2f a[4];
    const float* arow = rbf + (size_t)(e0 + l16) * R_DIM + 2 * half;
    #pragma unroll
    for (int s = 0; s < 4; ++s) {
        a[s].x = __builtin_nontemporal_load(arow + 4 * s + 0);
        a[s].y = __builtin_nontemporal_load(arow + 4 * s + 1);
    }

    // 32-bit element offsets into acc for the 8 D-rows this lane-half covers
    unsigned aoff[8];
    #pragma unroll
    for (int i = 0; i < 8; ++i)
        aoff[i] = (unsigned)idx1[e0 + i + 8 * half] * (unsigned)F_DIM + (unsigned)l16;

    // Single m_ji base: element (e0+8*half+i, 16t+l16) = base + i*128 + t*16
    const float* mbase = m_ji + (size_t)(e0 + 8 * half) * F_DIM + l16;

    #pragma unroll
    for (int t = 0; t < 8; ++t) {             // 8 feature tiles of 16
        v8f c = {};
        // B[K, N=feature]: B[k][n] = W_rbf[16t+n][k]; lane supplies column n=l16
        const float* brow = W_rbf + (16 * t + l16) * R_DIM + 2 * half;
        #pragma unroll
        for (int s = 0; s < 4; ++s) {
            v2f b;
            b.x = brow[4 * s + 0];
            b.y = brow[4 * s + 1];
            c = __builtin_amdgcn_wmma_f32_16x16x4_f32(
                    false, a[s], false, b, (short)0, c, false, false);
        }
        // batch the 8 independent gate loads first (single wait, overlapped)
        float mv[8];
        #pragma unroll
        for (int i = 0; i < 8; ++i)
            mv[i] = __builtin_nontemporal_load(mbase + i * F_DIM + t * 16);
        // then fire the 8 scatter atomics (no-return form, STOREcnt tracked)
        #pragma unroll
        for (int i = 0; i < 8; ++i)           // D: VGPR i -> row i + 8*half
            atomicAdd(acc + aoff[i] + t * 16, c[i] * mv[i]);
    }
}

// ---------------------------------------------------------------------------
// Kernel 2: fused 3-layer SiLU MLP + final 128->12 projection, all WMMA f32.
// One wave per 16-node tile; activations staged in LDS (A-layout transpose).
// ---------------------------------------------------------------------------
#define MLP_WAVES 4

__global__ __launch_bounds__(32 * MLP_WAVES) void mlp_out_kernel(
    const float* __restrict__ acc,     // [N_NODES,128]
    const float* __restrict__ dW,      // [3,128,128]  out[g] = sum_f x[f]*W[g][f]
    const float* __restrict__ db,      // [3,128]
    const float* __restrict__ Wout,    // [12,128]
    float*       __restrict__ out,     // [N_NODES,12]
    int nTiles)
{
    __shared__ float xs[MLP_WAVES][16 * F_DIM];   // 32 KB / block

    const int lane = threadIdx.x & 31;
    const int wave = threadIdx.x >> 5;
    const int l16  = lane & 15;
    const int half = lane >> 4;

    const int tile = blockIdx.x * MLP_WAVES + wave;
    if (tile >= nTiles) return;
    const unsigned n0 = (unsigned)tile * 16u;

    float* x = xs[wave];

    // cooperative load of this wave's 16x128 activation tile (coalesced rows)
    const float* arcv = acc + n0 * (unsigned)F_DIM + (unsigned)lane;
    #pragma unroll
    for (int r = 0; r < 16; ++r) {
        #pragma unroll
        for (int c0 = 0; c0 < F_DIM; c0 += 32)
            x[r * F_DIM + c0 + lane] = arcv[r * F_DIM + c0];
    }

    // ---- 3 dense layers: y = silu(x @ W^T + b) -----------------------------
    for (int L = 0; L < N_LAYER; ++L) {
        const float* WL = dW + (unsigned)L * F_DIM * F_DIM;
        float y[8][8];
        const float* axp = x + l16 * F_DIM + 2 * half;   // A: row=l16
        #pragma unroll
        for (int t = 0; t < 8; ++t) {
            v8f c = {};
            // B[k][n] = WL[16t+n][k] : lane n=l16 walks row (16t+l16) of WL
            const float* brow = WL + (16 * t + l16) * F_DIM + 2 * half;
            #pragma unroll
            for (int s = 0; s < 32; ++s) {               // K = 128 = 32 x 4
                v2f a, b;
                a.x = axp[4 * s + 0];  a.y = axp[4 * s + 1];
                b.x = brow[4 * s + 0]; b.y = brow[4 * s + 1];
                c = __builtin_amdgcn_wmma_f32_16x16x4_f32(
                        false, a, false, b, (short)0, c, false, false);
            }
            const float bias = db[L * F_DIM + 16 * t + l16];
            #pragma unroll
            for (int i = 0; i < 8; ++i) {
                const float v = c[i] + bias;
                y[t][i] = v / (1.0f + __expf(-v));       // SiLU
            }
        }
        // write results back to LDS in A-layout for the next layer
        #pragma unroll
        for (int t = 0; t < 8; ++t)
            #pragma unroll
            for (int i = 0; i < 8; ++i)
                x[(i + 8 * half) * F_DIM + 16 * t + l16] = y[t][i];
    }

    // ---- final projection: out[n,o] = sum_f x[n,f] * Wout[o][f], o < 12 ----
    {
        v8f c = {};
        const float* axp  = x + l16 * F_DIM + 2 * half;
        const int    brow_r = (l16 < O_DIM) ? l16 : 0;   // clamp, mask below
        const float  bmask  = (l16 < O_DIM) ? 1.0f : 0.0f;
        const float* brow = Wout + brow_r * F_DIM + 2 * half;
        #pragma unroll
        for (int s = 0; s < 32; ++s) {
            v2f a, b;
            a.x = axp[4 * s + 0];          a.y = axp[4 * s + 1];
            b.x = brow[4 * s + 0] * bmask; b.y = brow[4 * s + 1] * bmask;
            c = __builtin_amdgcn_wmma_f32_16x16x4_f32(
                    false, a, false, b, (short)0, c, false, false);
        }
        if (l16 < O_DIM) {
            float* op = out + (n0 + 8u * half) * (unsigned)O_DIM + (unsigned)l16;
            #pragma unroll
            for (int i = 0; i < 8; ++i)
                __builtin_nontemporal_store(c[i], op + i * O_DIM);
        }
    }
}

// ---------------------------------------------------------------------------
// Scalar tail for edge counts not divisible by 16 (not hit for E=640000)
// ---------------------------------------------------------------------------
__global__ void edge_tail_kernel(
    const float* __restrict__ m_ji, const float* __restrict__ rbf,
    const int* __restrict__ eidx, const float* __restrict__ W_rbf,
    float* __restrict__ acc, int E, int eStart)
{
    long g = (long)blockIdx.x * blockDim.x + threadIdx.x;
    long e = eStart + g / F_DIM;
    int  f = (int)(g % F_DIM);
    if (e >= E) return;
    float s = 0.0f;
    #pragma unroll
    for (int r = 0; r < R_DIM; ++r)
        s += rbf[e * R_DIM + r] * W_rbf[f * R_DIM + r];
    atomicAdd(&acc[(unsigned)eidx[E + e] * (unsigned)F_DIM + (unsigned)f],
              s * m_ji[e * F_DIM + f]);
}

// ---------------------------------------------------------------------------
extern "C" void kernel_launch(void* const* d_in, const int* in_sizes, int n_in,
                              void* d_out, int out_size, void* d_ws, size_t ws_size,
                              hipStream_t stream) {
    const float* m_ji = (const float*)d_in[0];
    const float* rbf  = (const float*)d_in[1];
    const int*   eidx = (const int*)  d_in[2];
    const float* Wrbf = (const float*)d_in[3];
    const float* dW   = (const float*)d_in[4];
    const float* db   = (const float*)d_in[5];
    const float* Wout = (const float*)d_in[6];
    float*       out  = (float*)d_out;

    const int E = in_sizes[0] / F_DIM;       // 640000
    float* acc = (float*)d_ws;               // [N_NODES,128] = 10.24 MB
    const long accN = (long)N_NODES * F_DIM;

    zero_acc_kernel<<<(int)((accN + 255) / 256), 256, 0, stream>>>(acc, accN);

    const int nEdgeTiles = E / 16;
    if (nEdgeTiles > 0)
        edge_gemm_scatter_kernel<<<(nEdgeTiles + 7) / 8, 256, 0, stream>>>(
            m_ji, rbf, eidx, Wrbf, acc, E);
    const int tail = E - nEdgeTiles * 16;
    if (tail > 0)
        edge_tail_kernel<<<(tail * F_DIM + 255) / 256, 256, 0, stream>>>(
            m_ji, rbf, eidx, Wrbf, acc, E, nEdgeTiles * 16);

    const int nNodeTiles = N_NODES / 16;     // 1250 exactly
    mlp_out_kernel<<<(nNodeTiles + MLP_WAVES - 1) / MLP_WAVES,
                     32 * MLP_WAVES, 0, stream>>>(acc, dW, db, Wout, out, nNodeTiles);
}